// QPGA_27470610825419
// MI455X (gfx1250) — compile-verified
//
#include <hip/hip_runtime.h>

// QPGA simulator, fused 8-layer pipeline, one batch element per block.
// bs = BS2^{⊗11} is Kronecker-factored:  v@bs = A16 · reshape(v,16,128) · B128,
// with A16 = BS2^{⊗4}, B128 = BS2^{⊗7}; entry(r,c) = scale * i^popcount(r^c).
// Both factor matmuls use V_WMMA_F32_16X16X4_F32 with the 3-product Gauss
// complex trick (Prr, Pii, Pss chains). Phase diagonals are separable:
// exp(i*theta(d)) = hi[d>>7] * lo[d&127], so only 16 + 128 sincos per apply per block.

#define NQ   11
#define DIM  2048
#define NL   8
#define NT   128          // 4 wave32s per block
#define WSTRIDE 130       // padded W row stride (floats) to avoid LDS bank conflicts

typedef __attribute__((ext_vector_type(2))) float v2f;
typedef __attribute__((ext_vector_type(8))) float v8f;

__device__ __forceinline__ v8f wmma4(v2f a, v2f b, v8f c) {
  // D = A(16x4) * B(4x16) + C, fp32 matrix core op (VOP3P V_WMMA_F32_16X16X4_F32)
  return __builtin_amdgcn_wmma_f32_16x16x4_f32(false, a, false, b, (short)0, c,
                                               false, false);
}

// Re / Im / (Re+Im) of i^pc, pc in 0..3
__device__ __forceinline__ float q_re(int pc) {
  return (pc == 0) ? 1.f : ((pc == 2) ? -1.f : 0.f);
}
__device__ __forceinline__ float q_im(int pc) {
  return (pc == 1) ? 1.f : ((pc == 3) ? -1.f : 0.f);
}
__device__ __forceinline__ float q_sum(int pc) {  // Re + Im
  return (pc < 2) ? 1.f : -1.f;
}

// state[d] *= exp(i * sum_q (bit_q(d) ? p1[q] : p0[q])) * (signParity>=0 ? cph : 1)
// Separable: qubits 0..3 live in bits 10..7 (= i), qubits 4..10 in bits 6..0 (= tid).
__device__ __forceinline__ void apply_phase(float* sR, float* sI, float* hiR,
                                            float* hiI,
                                            const float* __restrict__ p0,
                                            const float* __restrict__ p1,
                                            int signParity, int tid) {
  // hi-phasor table: 16 entries from qubits 0..3 (one sincos per entry)
  if (tid < 16) {
    float a = 0.f;
#pragma unroll
    for (int q = 0; q < 4; ++q) {
      const int sel = (tid >> (3 - q)) & 1;
      const float v1 = (p1 != nullptr) ? p1[q] : 0.f;
      a += sel ? v1 : p0[q];
    }
    float s, c;
    __sincosf(a, &s, &c);
    hiR[tid] = c;
    hiI[tid] = s;
  }
  // lo-phasor, per thread, from qubits 4..10 (bits 6..0 of tid)
  float la = 0.f;
#pragma unroll
  for (int q = 4; q < NQ; ++q) {
    const int sel = (tid >> (NQ - 1 - q)) & 1;
    const float v1 = (p1 != nullptr) ? p1[q] : 0.f;
    la += sel ? v1 : p0[q];
  }
  float ls, lc;
  __sincosf(la, &ls, &lc);
  __syncthreads();

  // CPHASE sign: -1^popc((d & ~(d<<1)) & mask); applied via float sign-bit XOR
  const int cmask = (signParity >= 0) ? (signParity ? 0x2AA : 0x554) : 0;

#pragma unroll
  for (int i = 0; i < DIM / NT; ++i) {
    const int d = i * NT + tid;
    float cs = hiR[i] * lc - hiI[i] * ls;
    float sn = hiR[i] * ls + hiI[i] * lc;
    if (signParity >= 0) {
      const int neg = (__popc((d & ~(d << 1)) & cmask) & 1) << 31;
      cs = __int_as_float(__float_as_int(cs) ^ neg);
      sn = __int_as_float(__float_as_int(sn) ^ neg);
    }
    const float r = sR[d], im = sI[d];
    sR[d] = r * cs - im * sn;
    sI[d] = r * sn + im * cs;
  }
  __syncthreads();
}

// One application of the beamsplitter mesh: S <- A16 * reshape(S,16,128) * B128.
// Fragment layouts (ISA 7.12.2, f32 16x16x4):
//   A 16x4 : lanes 0-15/16-31 = M,  VGPR0/1 = K {0,1} / {2,3}
//   B 4x16 : lanes = N, lane-half + VGPR select K (mirrors A)
//   C 16x16: VGPR v -> M = v (+8 for upper lanes), lane = N
__device__ __forceinline__ void bs_apply(float* sR, float* sI, float* wR,
                                         float* wI, float* wS, int tid) {
  const int lane  = tid & 31;
  // Wave id as an SGPR: scalar tile control flow, WMMA always runs full-EXEC.
  const int wave  = __builtin_amdgcn_readfirstlane(tid >> 5);
  const int li    = lane & 15;          // M (A-frag) or N (B/C-frag)
  const int kHalf = (lane >> 4) << 1;   // +0 or +2 on K
  const int mHi   = (lane >> 4) << 3;   // +0 or +8 on C-layout M
  const float S4 = 0.25f;                // (1/sqrt2)^4
  const float S7 = 0.08838834764831845f; // (1/sqrt2)^7

  // ---- Step 1: W(16x128) = A16 * V,  V[k][n] = state[k*128 + n] ----
  for (int t = wave; t < 8; t += 4) {
    const int n0 = t * 16;
    v8f prr = {}, pii = {}, pss = {};
#pragma unroll
    for (int kk = 0; kk < 4; ++kk) {
      const int k0 = kk * 4 + kHalf;
      // A16 fragments (compile-time-foldable constants per lane)
      const int p0 = __popc(li ^ k0) & 3;
      const int p1 = __popc(li ^ (k0 + 1)) & 3;
      v2f ar, ai, as;
      ar.x = S4 * q_re(p0);  ar.y = S4 * q_re(p1);
      ai.x = S4 * q_im(p0);  ai.y = S4 * q_im(p1);
      as.x = S4 * q_sum(p0); as.y = S4 * q_sum(p1);
      // V fragments from LDS
      const int n = n0 + li;
      v2f br, bi, bs;
      br.x = sR[k0 * 128 + n];  br.y = sR[(k0 + 1) * 128 + n];
      bi.x = sI[k0 * 128 + n];  bi.y = sI[(k0 + 1) * 128 + n];
      bs.x = br.x + bi.x;       bs.y = br.y + bi.y;
      prr = wmma4(ar, br, prr);
      pii = wmma4(ai, bi, pii);
      pss = wmma4(as, bs, pss);
    }
#pragma unroll
    for (int v = 0; v < 8; ++v) {
      const int m = v + mHi;
      const int n = n0 + li;
      wR[m * WSTRIDE + n] = prr[v] - pii[v];            // Re
      wI[m * WSTRIDE + n] = pss[v] - prr[v] - pii[v];   // Im
      wS[m * WSTRIDE + n] = pss[v] - 2.f * pii[v];      // Re + Im
    }
  }
  __syncthreads();

  // ---- Step 2: S = W * B128,  B128[k][n] = S7 * i^popc(k^n) ----
  // Both tiles of this wave share the W fragments (A-operand depends only on
  // kk), so process them together: one LDS fragment load feeds 6 WMMAs.
  {
    const int n0a = wave * 16;       // tile t = wave
    const int n0b = n0a + 64;        // tile t = wave + 4
    v8f prrA = {}, piiA = {}, pssA = {};
    v8f prrB = {}, piiB = {}, pssB = {};
    for (int kk = 0; kk < 32; ++kk) {
      const int k0 = kk * 4 + kHalf;
      // W fragments from LDS (padded stride dodges bank conflicts)
      v2f ar, ai, as;
      ar.x = wR[li * WSTRIDE + k0];  ar.y = wR[li * WSTRIDE + k0 + 1];
      ai.x = wI[li * WSTRIDE + k0];  ai.y = wI[li * WSTRIDE + k0 + 1];
      as.x = wS[li * WSTRIDE + k0];  as.y = wS[li * WSTRIDE + k0 + 1];
      // B128 fragments, generated in-register, for both tiles
      const int nA = n0a + li;
      const int nB = n0b + li;
      const int pa0 = __popc(k0 ^ nA) & 3;
      const int pa1 = __popc((k0 + 1) ^ nA) & 3;
      const int pb0 = __popc(k0 ^ nB) & 3;
      const int pb1 = __popc((k0 + 1) ^ nB) & 3;
      v2f brA, biA, bsA, brB, biB, bsB;
      brA.x = S7 * q_re(pa0);  brA.y = S7 * q_re(pa1);
      biA.x = S7 * q_im(pa0);  biA.y = S7 * q_im(pa1);
      bsA.x = S7 * q_sum(pa0); bsA.y = S7 * q_sum(pa1);
      brB.x = S7 * q_re(pb0);  brB.y = S7 * q_re(pb1);
      biB.x = S7 * q_im(pb0);  biB.y = S7 * q_im(pb1);
      bsB.x = S7 * q_sum(pb0); bsB.y = S7 * q_sum(pb1);
      prrA = wmma4(ar, brA, prrA);
      piiA = wmma4(ai, biA, piiA);
      pssA = wmma4(as, bsA, pssA);
      prrB = wmma4(ar, brB, prrB);
      piiB = wmma4(ai, biB, piiB);
      pssB = wmma4(as, bsB, pssB);
    }
#pragma unroll
    for (int v = 0; v < 8; ++v) {
      const int m = v + mHi;
      sR[m * 128 + n0a + li] = prrA[v] - piiA[v];
      sI[m * 128 + n0a + li] = pssA[v] - prrA[v] - piiA[v];
      sR[m * 128 + n0b + li] = prrB[v] - piiB[v];
      sI[m * 128 + n0b + li] = pssB[v] - prrB[v] - piiB[v];
    }
  }
  __syncthreads();
}

__global__ __launch_bounds__(NT) void qpga_kernel(
    const float* __restrict__ x, const float* __restrict__ alphas,
    const float* __restrict__ betas, const float* __restrict__ thetas,
    const float* __restrict__ phis, float* __restrict__ out) {
  __shared__ float sR[DIM];
  __shared__ float sI[DIM];
  __shared__ float wR[16 * WSTRIDE];
  __shared__ float wI[16 * WSTRIDE];
  __shared__ float wS[16 * WSTRIDE];
  __shared__ float hiR[16];
  __shared__ float hiI[16];

  const int tid = threadIdx.x;
  const size_t base = (size_t)blockIdx.x * 2 * DIM;
  const float* xr = x + base;
  const float* xi = xr + DIM;

#pragma unroll
  for (int i = 0; i < DIM / NT; ++i) {
    const int d = i * NT + tid;
    sR[d] = xr[d];
    sI[d] = xi[d];
  }
  __syncthreads();

  for (int l = 0; l < NL; ++l) {
    apply_phase(sR, sI, hiR, hiI, alphas + l * NQ, betas + l * NQ, -1, tid);
    bs_apply(sR, sI, wR, wI, wS, tid);
    apply_phase(sR, sI, hiR, hiI, thetas + l * NQ, nullptr, -1, tid);
    bs_apply(sR, sI, wR, wI, wS, tid);
    apply_phase(sR, sI, hiR, hiI, phis + l * NQ, nullptr, l & 1, tid);
  }

  float* outr = out + base;
  float* outi = outr + DIM;
#pragma unroll
  for (int i = 0; i < DIM / NT; ++i) {
    const int d = i * NT + tid;
    outr[d] = sR[d];
    outi[d] = sI[d];
  }
}

extern "C" void kernel_launch(void* const* d_in, const int* in_sizes, int n_in,
                              void* d_out, int out_size, void* d_ws,
                              size_t ws_size, hipStream_t stream) {
  const float* x      = (const float*)d_in[0];
  const float* alphas = (const float*)d_in[1];
  const float* betas  = (const float*)d_in[2];
  const float* thetas = (const float*)d_in[3];
  const float* phis   = (const float*)d_in[4];
  float* out = (float*)d_out;

  const int batch = in_sizes[0] / (2 * DIM); // 1024
  qpga_kernel<<<batch, NT, 0, stream>>>(x, alphas, betas, thetas, phis, out);

  (void)n_in; (void)out_size; (void)d_ws; (void)ws_size;
}